// RRMamba_Forecast_51230369907308
// MI455X (gfx1250) — compile-verified
//
#include <hip/hip_runtime.h>
#include <hip/hip_bf16.h>

// ---------------------------------------------------------------------------
// RRMamba forecast, CDNA5 (gfx1250).
// H=128, NL=4, ED=256, N=16, DTR=8, K=4, B=16, L=1024, TOK=16384.
// GEMMs: V_WMMA_F32_16X16X4_F32, wave computes a 16x(16*NT) tile, K templated
// for full strength reduction. Scan: states in VGPRs, prefetch lookahead.
// ---------------------------------------------------------------------------

typedef __attribute__((ext_vector_type(2))) float v2f;
typedef __attribute__((ext_vector_type(8))) float v8f;

#define TOKC 16384
#define WAVES 8

__device__ __forceinline__ float silu_f(float v) {
    return v / (1.0f + __expf(-v));
}

// ---------------------------------------------------------------------------
// WMMA f32 GEMM:  C[M,Npad] = A[M,KCT] @ W[KCT,N]  (+ fused epilogue)
// One wave -> 16 x (16*NT) output tile (NT accumulators share the A frag).
// grid = (Npad/(16*NT), M/128), block = 256 (8 waves).
// EPI: 0 = store | 1 = +bias,SiLU | 2 = +bias,+aux[batch][col],SiLU
//      3 = +bias,softplus | 4 = accumulate into C
// NGUARD: mask B columns >= Nreal (only needed for the N=40 xproj).
// ---------------------------------------------------------------------------
template <int EPI, int KCT, int NT, bool NGUARD>
__global__ __launch_bounds__(256) void k_gemm_wmma(
    const float* __restrict__ A, int lda,
    const float* __restrict__ W, int ldw, int Nreal,
    float* __restrict__ C, int ldc,
    const float* __restrict__ bias,
    const float* __restrict__ aux)
{
    const int lane = threadIdx.x & 31;
    const int wave = threadIdx.x >> 5;
    const int m0 = (blockIdx.y * WAVES + wave) * 16;
    const int n0 = blockIdx.x * (16 * NT);
    const int half = lane >> 4;      // 0: lanes 0-15, 1: lanes 16-31
    const int l16  = lane & 15;
    const int kp   = half * 2;       // K offset within the 4-wide slab
    const int arow = m0 + l16;

    int   bcol[NT], bcc[NT];
    float bmsk[NT];
#pragma unroll
    for (int t = 0; t < NT; ++t) {
        bcol[t] = n0 + 16 * t + l16;
        if (NGUARD) {
            const bool ok = bcol[t] < Nreal;
            bcc[t]  = ok ? bcol[t] : 0;
            bmsk[t] = ok ? 1.0f : 0.0f;
        } else {
            bcc[t]  = bcol[t];
            bmsk[t] = 1.0f;
        }
    }

    v8f acc[NT];
#pragma unroll
    for (int t = 0; t < NT; ++t) acc[t] = (v8f){};

    const float* ap = A + (size_t)arow * lda + kp;
    const float* bp = W + (size_t)kp * ldw;

#pragma unroll 8
    for (int k0 = 0; k0 < KCT; k0 += 4) {
        v2f a;                       // A[arow][k0+kp .. +1] -> one b64 load
        a[0] = ap[0];
        a[1] = ap[1];
        ap += 4;
#pragma unroll
        for (int t = 0; t < NT; ++t) {
            v2f b;                   // W[k0+kp][col], W[k0+kp+1][col]
            if (NGUARD) {
                b[0] = bp[bcc[t]] * bmsk[t];
                b[1] = bp[(size_t)ldw + bcc[t]] * bmsk[t];
            } else {
                b[0] = bp[bcc[t]];
                b[1] = bp[(size_t)ldw + bcc[t]];
            }
            acc[t] = __builtin_amdgcn_wmma_f32_16x16x4_f32(
                false, a, false, b, (short)0, acc[t], false, false);
        }
        bp += (size_t)4 * ldw;
    }

    // C/D layout: VGPR j -> rows m0+j (lanes 0-15) and m0+j+8 (lanes 16-31)
#pragma unroll
    for (int j = 0; j < 8; ++j) {
        const int row = m0 + j + half * 8;
#pragma unroll
        for (int t = 0; t < NT; ++t) {
            float v = acc[t][j];
            if (EPI == 1 || EPI == 2 || EPI == 3) v += bias[bcc[t]];
            if (EPI == 2) v += aux[(row >> 10) * 128 + bcc[t]];  // row>>10 = batch
            if (EPI == 1 || EPI == 2) v = silu_f(v);
            if (EPI == 3) v = (v > 20.0f) ? v : log1pf(__expf(v));
            if (EPI == 4) v += C[(size_t)row * ldc + bcol[t]];
            C[(size_t)row * ldc + bcol[t]] = v;
        }
    }
}

// ---------------------------------------------------------------------------
// Static branch: s = silu(static @ stat_W + stat_b) [16x128], then
// sOut = s @ comb_W1[128:256,:]  [16x128]. Tiny, single block.
// ---------------------------------------------------------------------------
__global__ __launch_bounds__(256) void k_stat(
    const float* __restrict__ stat_in, const float* __restrict__ statW,
    const float* __restrict__ statb, const float* __restrict__ combW1,
    float* __restrict__ sOut)
{
    __shared__ float s[16 * 128];
    for (int o = threadIdx.x; o < 2048; o += 256) {
        const int b = o >> 7, c = o & 127;
        float a = statb[c];
#pragma unroll
        for (int j = 0; j < 16; ++j) a += stat_in[b * 16 + j] * statW[j * 128 + c];
        s[o] = silu_f(a);
    }
    __syncthreads();
    for (int o = threadIdx.x; o < 2048; o += 256) {
        const int b = o >> 7, c = o & 127;
        float a = 0.0f;
        for (int j = 0; j < 128; ++j) a += s[b * 128 + j] * combW1[(128 + j) * 128 + c];
        sOut[o] = a;
    }
}

// ---------------------------------------------------------------------------
// RMSNorm: one wave per row of 128. 8 rows per block.
// ---------------------------------------------------------------------------
__global__ __launch_bounds__(256) void k_rmsnorm(
    const float* __restrict__ h, const float* __restrict__ w,
    float* __restrict__ o)
{
    const int lane = threadIdx.x & 31;
    const int wave = threadIdx.x >> 5;
    const int row  = blockIdx.x * 8 + wave;
    const float* hr = h + (size_t)row * 128;
    float v0 = hr[lane], v1 = hr[lane + 32], v2 = hr[lane + 64], v3 = hr[lane + 96];
    float ss = v0 * v0 + v1 * v1 + v2 * v2 + v3 * v3;
#pragma unroll
    for (int off = 16; off >= 1; off >>= 1) ss += __shfl_xor(ss, off, 32);
    const float r = rsqrtf(ss * (1.0f / 128.0f) + 1e-5f);
    float* orow = o + (size_t)row * 128;
    orow[lane]      = v0 * r * w[lane];
    orow[lane + 32] = v1 * r * w[lane + 32];
    orow[lane + 64] = v2 * r * w[lane + 64];
    orow[lane + 96] = v3 * r * w[lane + 96];
}

// ---------------------------------------------------------------------------
// Causal depthwise conv (K=4) over xb = xz[:, :256], +bias, SiLU -> xbc.
// Also zs = silu(xz[:, 256:512]). One thread per (token, channel).
// ---------------------------------------------------------------------------
__global__ __launch_bounds__(256) void k_conv(
    const float* __restrict__ xz, const float* __restrict__ cw,
    const float* __restrict__ cb, float* __restrict__ xbc,
    float* __restrict__ zs)
{
    const int idx = blockIdx.x * 256 + threadIdx.x;   // over TOK*256
    const int t = idx >> 8;
    const int e = idx & 255;
    const int l = t & 1023;
    float acc = cb[e];
    const float* w4 = cw + e * 4;
#pragma unroll
    for (int k = 0; k < 4; ++k) {
        const int lm = l + k - 3;
        if (lm >= 0) acc += xz[(size_t)(t + k - 3) * 512 + e] * w4[k];
    }
    xbc[idx] = silu_f(acc);
    const float z = xz[(size_t)t * 512 + 256 + e];
    zs[idx] = silu_f(z);
}

// ---------------------------------------------------------------------------
// Selective scan. One block (256 threads) per batch element; thread e owns
// channel e with all N=16 recurrent states in VGPRs -> no barriers in the
// 1024-step loop. Bm/Cm rows fetched as float4 (b128) broadcast loads;
// global_prefetch_b8 lookahead hides the sequential-load latency.
// dbl row layout (ldc=48): [0:8]=dt_low, [8:24]=Bm, [24:40]=Cm.
// ---------------------------------------------------------------------------
__global__ __launch_bounds__(256) void k_scan(
    const float* __restrict__ delta, const float* __restrict__ xbc,
    const float* __restrict__ dbl, const float* __restrict__ zs,
    const float* __restrict__ Alog, const float* __restrict__ Dp,
    float* __restrict__ y)
{
    const int e = threadIdx.x;
    const int b = blockIdx.x;
    float Arow[16];
#pragma unroll
    for (int n = 0; n < 16; ++n) Arow[n] = -__expf(Alog[e * 16 + n]);
    const float dpe = Dp[e];
    float st[16];
#pragma unroll
    for (int n = 0; n < 16; ++n) st[n] = 0.0f;

    const int t0 = b * 1024;
    for (int l = 0; l < 1024; ++l) {
        const int t = t0 + l;
        if (l + 8 < 1024) {  // emits global_prefetch_b8 (gfx1250)
            __builtin_prefetch(delta + (size_t)(t + 8) * 256 + e, 0, 0);
            __builtin_prefetch(xbc   + (size_t)(t + 8) * 256 + e, 0, 0);
            __builtin_prefetch(dbl   + (size_t)(t + 8) * 48, 0, 0);
        }
        const float de = delta[(size_t)t * 256 + e];
        const float xe = xbc[(size_t)t * 256 + e];
        const float ze = zs[(size_t)t * 256 + e];

        // Bm (16 floats) + Cm (16 floats): 8 x b128 broadcast loads.
        float bv[16], cvv[16];
        const float4* p4 = reinterpret_cast<const float4*>(dbl + (size_t)t * 48 + 8);
#pragma unroll
        for (int g = 0; g < 4; ++g) {
            const float4 q = p4[g];
            bv[4 * g + 0] = q.x; bv[4 * g + 1] = q.y;
            bv[4 * g + 2] = q.z; bv[4 * g + 3] = q.w;
        }
#pragma unroll
        for (int g = 0; g < 4; ++g) {
            const float4 q = p4[4 + g];
            cvv[4 * g + 0] = q.x; cvv[4 * g + 1] = q.y;
            cvv[4 * g + 2] = q.z; cvv[4 * g + 3] = q.w;
        }

        const float du = de * xe;
        float acc = 0.0f;
#pragma unroll
        for (int n = 0; n < 16; ++n) {
            const float dA = __expf(de * Arow[n]);
            st[n] = fmaf(dA, st[n], du * bv[n]);
            acc = fmaf(st[n], cvv[n], acc);
        }
        y[(size_t)t * 256 + e] = (acc + dpe * xe) * ze;
    }
}

// ---------------------------------------------------------------------------
// Final head: only the last token per batch. out[b] = silu(h_last) @ out_W + b
// ---------------------------------------------------------------------------
__global__ __launch_bounds__(512) void k_final(
    const float* __restrict__ h, const float* __restrict__ oW,
    const float* __restrict__ ob, float* __restrict__ out)
{
    const int lane = threadIdx.x & 31;
    const int b = threadIdx.x >> 5;
    const float* hr = h + ((size_t)b * 1024 + 1023) * 128;
    float acc = 0.0f;
#pragma unroll
    for (int j = 0; j < 4; ++j) {
        const int c = lane + 32 * j;
        acc += silu_f(hr[c]) * oW[c];
    }
#pragma unroll
    for (int off = 16; off >= 1; off >>= 1) acc += __shfl_xor(acc, off, 32);
    if (lane == 0) out[b] = acc + ob[0];
}

// ---------------------------------------------------------------------------
extern "C" void kernel_launch(void* const* d_in, const int* in_sizes, int n_in,
                              void* d_out, int out_size, void* d_ws, size_t ws_size,
                              hipStream_t stream)
{
    (void)in_sizes; (void)n_in; (void)out_size; (void)ws_size;
    const float* x        = (const float*)d_in[0];
    const float* stat_in  = (const float*)d_in[1];
    const float* emb_W    = (const float*)d_in[2];
    const float* emb_b    = (const float*)d_in[3];
    const float* stat_W   = (const float*)d_in[4];
    const float* stat_b   = (const float*)d_in[5];
    const float* comb_W1  = (const float*)d_in[6];
    const float* comb_b1  = (const float*)d_in[7];
    const float* comb_W2  = (const float*)d_in[8];
    const float* comb_b2  = (const float*)d_in[9];
    const float* norm_w   = (const float*)d_in[10];
    const float* in_proj  = (const float*)d_in[11];
    const float* conv_W   = (const float*)d_in[12];
    const float* conv_b   = (const float*)d_in[13];
    const float* xproj_W  = (const float*)d_in[14];
    const float* dtproj_W = (const float*)d_in[15];
    const float* dtproj_b = (const float*)d_in[16];
    const float* A_log    = (const float*)d_in[17];
    const float* Dp       = (const float*)d_in[18];
    const float* outproj  = (const float*)d_in[19];
    const float* out_W    = (const float*)d_in[20];
    const float* out_b    = (const float*)d_in[21];
    float* out = (float*)d_out;

    // Workspace carve-up (floats), ~121 MB total.
    float* ws    = (float*)d_ws;
    float* h     = ws;                          // TOK*128
    float* hn    = h     + (size_t)TOKC * 128;
    float* xz    = hn    + (size_t)TOKC * 128;  // TOK*512
    float* xbc   = xz    + (size_t)TOKC * 512;  // TOK*256 (front-end reuses as TOK*128)
    float* zs    = xbc   + (size_t)TOKC * 256;
    float* dbl   = zs    + (size_t)TOKC * 256;  // TOK*48 (padded from 40)
    float* delta = dbl   + (size_t)TOKC * 48;
    float* yb    = delta + (size_t)TOKC * 256;
    float* sW    = yb    + (size_t)TOKC * 256;  // 16*128

    const dim3 blk(256);
    const int MBLK = TOKC / (16 * WAVES);       // 128 blocks along M

    // Static branch contribution to comb1.
    k_stat<<<1, 256, 0, stream>>>(stat_in, stat_W, stat_b, comb_W1, sW);

    // h1 = silu(x @ emb_W + emb_b)                        -> hn
    k_gemm_wmma<1, 32, 2, false><<<dim3(128 / 32, MBLK), blk, 0, stream>>>(
        x, 32, emb_W, 128, 128, hn, 128, emb_b, nullptr);
    // t = silu(h1 @ comb_W1[:128] + sW[batch] + comb_b1)  -> xbc (as TOK x 128)
    k_gemm_wmma<2, 128, 2, false><<<dim3(128 / 32, MBLK), blk, 0, stream>>>(
        hn, 128, comb_W1, 128, 128, xbc, 128, comb_b1, sW);
    // h = silu(t @ comb_W2 + comb_b2)                     -> h
    k_gemm_wmma<1, 128, 2, false><<<dim3(128 / 32, MBLK), blk, 0, stream>>>(
        xbc, 128, comb_W2, 128, 128, h, 128, comb_b2, nullptr);

    for (int i = 0; i < 4; ++i) {
        // hn = rmsnorm(h, norm_w[i])
        k_rmsnorm<<<TOKC / 8, 256, 0, stream>>>(h, norm_w + i * 128, hn);
        // xz = hn @ in_proj_W[i]   (TOK x 512)
        k_gemm_wmma<0, 128, 2, false><<<dim3(512 / 32, MBLK), blk, 0, stream>>>(
            hn, 128, in_proj + (size_t)i * 128 * 512, 512, 512, xz, 512,
            nullptr, nullptr);
        // xbc = silu(conv(xz[:, :256]) + conv_b), zs = silu(xz[:, 256:])
        k_conv<<<TOKC, 256, 0, stream>>>(
            xz, conv_W + (size_t)i * 256 * 4, conv_b + (size_t)i * 256, xbc, zs);
        // dbl = xbc @ xproj_W[i]   (TOK x 40, padded to 48; guarded cols)
        k_gemm_wmma<0, 256, 1, true><<<dim3(48 / 16, MBLK), blk, 0, stream>>>(
            xbc, 256, xproj_W + (size_t)i * 256 * 40, 40, 40, dbl, 48,
            nullptr, nullptr);
        // delta = softplus(dbl[:, :8] @ dtproj_W[i] + dtproj_b[i])
        k_gemm_wmma<3, 8, 2, false><<<dim3(256 / 32, MBLK), blk, 0, stream>>>(
            dbl, 48, dtproj_W + (size_t)i * 8 * 256, 256, 256, delta, 256,
            dtproj_b + (size_t)i * 256, nullptr);
        // selective scan -> y
        k_scan<<<16, 256, 0, stream>>>(
            delta, xbc, dbl, zs, A_log + (size_t)i * 256 * 16,
            Dp + (size_t)i * 256, yb);
        // h += y @ outproj_W[i]
        k_gemm_wmma<4, 256, 2, false><<<dim3(128 / 32, MBLK), blk, 0, stream>>>(
            yb, 256, outproj + (size_t)i * 256 * 128, 128, 128, h, 128,
            nullptr, nullptr);
    }

    k_final<<<1, 512, 0, stream>>>(h, out_W, out_b, out);
}